// DoublyEquivariantOrbitalLayer_16767552323752
// MI455X (gfx1250) — compile-verified
//
#include <hip/hip_runtime.h>

typedef __attribute__((ext_vector_type(2))) float v2f;
typedef __attribute__((ext_vector_type(8))) float v8f;

// Dims: B=1024, NELEC=64, NSPIN=2, NE=32, D=32, NION=16, SPATIAL=3, NORB=32
#define NB    1024
#define NE    32
#define DD    32
#define NION  16
#define NORB  32

__global__ __launch_bounds__(256) void deq_orbital_kernel(
    const float* __restrict__ x,       // (B, 64, 32)
    const float* __restrict__ r_ei,    // (B, 64, 16, 3)
    const float* __restrict__ W_orb,   // (2, 64, 32)
    const float* __restrict__ b_orb,   // (2, 32)
    const float* __restrict__ W_env,   // (2, 16, 3, 32, 3)
    const float* __restrict__ w_ion,   // (2, 16)
    float* __restrict__ out)           // (2, B, 32, 32, 32)
{
    __shared__ float Xs[NE * DD];            // 32x32 electron features
    __shared__ float Wo[2 * DD * NORB];      // 64x32 orbital weights
    __shared__ float Bo[NORB];
    __shared__ float Rr[NE * NION * 3];      // 32x48 electron-ion vectors
    __shared__ float We[NION * 3 * NORB * 3];// 4608 env matrices
    __shared__ float Wi[NION];
    __shared__ float Ut[NE * NORB];          // U[p,o]
    __shared__ float Vt[NE * NORB];          // V[i,o] + bias
    __shared__ float Ev[NE * NORB];          // env[i,o]

    const int tid = threadIdx.x;
    const int sb  = blockIdx.x;              // s*B + b
    const int s   = sb >> 10;
    const int b   = sb & 1023;

    // ---------------- cooperative global -> LDS loads (float4, coalesced) ----
    {
        const float4* xsrc = (const float4*)(x + (size_t)b * 2048 + (size_t)s * 1024);
        ((float4*)Xs)[tid] = xsrc[tid];                        // 1024 floats

        const float4* wsrc = (const float4*)(W_orb + (size_t)s * 2048);
        ((float4*)Wo)[tid]       = wsrc[tid];
        ((float4*)Wo)[tid + 256] = wsrc[tid + 256];            // 2048 floats

        if (tid < 32) Bo[tid] = b_orb[s * 32 + tid];

        const float4* rsrc = (const float4*)(r_ei + (size_t)b * 3072 + (size_t)s * 1536);
        ((float4*)Rr)[tid] = rsrc[tid];
        if (tid < 128) ((float4*)Rr)[tid + 256] = rsrc[tid + 256];   // 1536 floats

        const float4* esrc = (const float4*)(W_env + (size_t)s * 4608);
        for (int i = tid; i < 1152; i += 256) ((float4*)We)[i] = esrc[i]; // 4608 floats

        if (tid < 16) Wi[tid] = w_ion[s * 16 + tid];
    }
    __syncthreads();

    // ---------------- WMMA: U = X @ Worb[:32], V = X @ Worb[32:] (+bias) ------
    // waves 0-3 -> four 16x16 tiles of U; waves 4-7 -> four tiles of V
    {
        const int wave = tid >> 5;
        const int lane = tid & 31;
        const int isV  = wave >> 2;
        const int w4   = wave & 3;
        const int pm   = w4 >> 1;               // tile row
        const int po   = w4 & 1;                // tile col
        const int mrow = pm * 16 + (lane & 15); // A-matrix row (M)
        const int ncol = po * 16 + (lane & 15); // B-matrix col (N)
        const int khi  = lane >> 4;             // lane-half selects K phase
        const int foff = isV * 32;

        v8f c = {};
        #pragma unroll
        for (int kk = 0; kk < 8; ++kk) {
            const int kb = kk * 4 + 2 * khi;    // VGPR0:K=kb, VGPR1:K=kb+1
            v2f a, bb;
            a.x  = Xs[mrow * 32 + kb];
            a.y  = Xs[mrow * 32 + kb + 1];
            bb.x = Wo[(foff + kb    ) * 32 + ncol];
            bb.y = Wo[(foff + kb + 1) * 32 + ncol];
            c = __builtin_amdgcn_wmma_f32_16x16x4_f32(false, a, false, bb,
                                                      (short)0, c, false, false);
        }
        const float bias = isV ? Bo[ncol] : 0.0f;
        float* dst = isV ? Vt : Ut;
        #pragma unroll
        for (int r = 0; r < 8; ++r) {
            const int m = r + 8 * khi;          // D layout: VGPRr -> M=r / M=r+8
            dst[(pm * 16 + m) * 32 + ncol] = c[r] + bias;
        }
    }

    // ---------------- anisotropic envelope: env[n,o] = sum_i exp(-|A r|) w_i --
    {
        const int n     = tid >> 3;             // electron row, fixed per thread
        const int obase = (tid & 7) * 4;        // 4 orbitals per thread
        float acc0 = 0.f, acc1 = 0.f, acc2 = 0.f, acc3 = 0.f;
        for (int i = 0; i < NION; ++i) {
            const float r0 = Rr[n * 48 + i * 3 + 0];
            const float r1 = Rr[n * 48 + i * 3 + 1];
            const float r2 = Rr[n * 48 + i * 3 + 2];
            const float wi = Wi[i];
            #pragma unroll
            for (int e4 = 0; e4 < 4; ++e4) {
                // We[((i*3+d)*32+o)*3+e] = We[i*288 + d*96 + o*3 + e]
                const float* A = &We[i * 288 + (obase + e4) * 3];
                const float s0 = fmaf(r0, A[0], fmaf(r1, A[96], r2 * A[192]));
                const float s1 = fmaf(r0, A[1], fmaf(r1, A[97], r2 * A[193]));
                const float s2 = fmaf(r0, A[2], fmaf(r1, A[98], r2 * A[194]));
                const float dist = sqrtf(fmaf(s0, s0, fmaf(s1, s1, s2 * s2)));
                const float ev = __expf(-dist) * wi;
                if      (e4 == 0) acc0 += ev;
                else if (e4 == 1) acc1 += ev;
                else if (e4 == 2) acc2 += ev;
                else              acc3 += ev;
            }
        }
        Ev[n * 32 + obase + 0] = acc0;
        Ev[n * 32 + obase + 1] = acc1;
        Ev[n * 32 + obase + 2] = acc2;
        Ev[n * 32 + obase + 3] = acc3;
    }
    __syncthreads();

    // ---------------- streaming output: out[p,i,o] = (U[p,o]+Vb[i,o])*env[i,o]
    {
        float4* o4p = (float4*)(out + (size_t)sb * 32768);
        #pragma unroll 4
        for (int it = 0; it < 32; ++it) {
            const int idx4 = it * 256 + tid;    // 0..8191 float4s
            const int o4   = (idx4 & 7) * 4;
            const int i    = (idx4 >> 3) & 31;
            const int p    = idx4 >> 8;
            const float4 u = *(const float4*)&Ut[p * 32 + o4];
            const float4 v = *(const float4*)&Vt[i * 32 + o4];
            const float4 e = *(const float4*)&Ev[i * 32 + o4];
            float4 r;
            r.x = (u.x + v.x) * e.x;
            r.y = (u.y + v.y) * e.y;
            r.z = (u.z + v.z) * e.z;
            r.w = (u.w + v.w) * e.w;
            o4p[idx4] = r;                      // coalesced global_store_b128
        }
    }
}

extern "C" void kernel_launch(void* const* d_in, const int* in_sizes, int n_in,
                              void* d_out, int out_size, void* d_ws, size_t ws_size,
                              hipStream_t stream) {
    const float* x     = (const float*)d_in[0];
    const float* r_ei  = (const float*)d_in[1];
    const float* W_orb = (const float*)d_in[2];
    const float* b_orb = (const float*)d_in[3];
    const float* W_env = (const float*)d_in[4];
    const float* w_ion = (const float*)d_in[5];
    float* out = (float*)d_out;

    deq_orbital_kernel<<<dim3(2 * NB), dim3(256), 0, stream>>>(
        x, r_ei, W_orb, b_orb, W_env, w_ion, out);
}